// FeaturePropagation_89438398972533
// MI455X (gfx1250) — compile-verified
//
#include <hip/hip_runtime.h>
#include <hip/hip_bf16.h>

// ---------------------------------------------------------------------------
// FeaturePropagation (PointNet++ FP): knn_interpolate(k=3) -> concat ->
// relu(h@W1+b1) -> relu(h@W2+b2).  B=8, N_SRC=2048, N_DST=8192, C=256,
// C_SKIP=64.  M = 65536 rows total.
//
// Kernel 1: brute-force KNN (src pos staged in LDS), emits normalized
//           inverse-d2 weights + global source row indices (1.5 MB ws).
// Kernel 2: fully fused per-16-row tile: interpolate+concat into LDS (A0),
//           GEMM1 (K=320) via V_WMMA_F32_16X16X4_F32, bias+ReLU into LDS,
//           GEMM2 (K=256) via WMMA, bias+ReLU to d_out. No h0/h1 in HBM.
// ---------------------------------------------------------------------------

#define NB      8
#define NSRC    2048
#define NDST    8192
#define CH      256
#define CSKIP   64
#define MTOT    (NB * NDST)      // 65536
#define K1      (CH + CSKIP)     // 320
#define NOUT    256
#define LDA     324              // 320 cols + pad; 324 % 64 == 4 -> conflict-free col reads
#define LDH     260              // 256 cols + pad; 260 % 64 == 4

typedef __attribute__((ext_vector_type(2))) float v2f;
typedef __attribute__((ext_vector_type(8))) float v8f;

static __device__ __forceinline__ v8f wmma_f32_4(v2f a, v2f b, v8f c) {
    // D = A(16x4,f32) * B(4x16,f32) + C(16x16,f32)
    return __builtin_amdgcn_wmma_f32_16x16x4_f32(
        /*neg_a=*/false, a, /*neg_b=*/false, b,
        /*c_mod=*/(short)0, c, /*reuse_a=*/false, /*reuse_b=*/false);
}

// ---------------------------------------------------------------------------
// Kernel 1: KNN (k=3) + inverse-squared-distance weights (normalized)
// grid = MTOT/256 blocks of 256 threads; 32 blocks per cloud.
// ---------------------------------------------------------------------------
__global__ __launch_bounds__(256) void fp_knn_kernel(
    const float* __restrict__ pos,        // [B*NSRC, 3]
    const float* __restrict__ pos_skip,   // [B*NDST, 3]
    int*   __restrict__ g_idx,            // [MTOT, 3] global src row index
    float* __restrict__ g_w)              // [MTOT, 3] normalized weights
{
    __shared__ float4 sp[NSRC];           // 32 KB: padded xyz of this cloud's sources

    const int t       = threadIdx.x;
    const int cloud   = blockIdx.x >> 5;          // 8192/256 = 32 blocks per cloud
    const int srcbase = cloud * NSRC;

    for (int p = t; p < NSRC; p += 256) {
        const float* q = pos + (size_t)(srcbase + p) * 3;
        sp[p] = make_float4(q[0], q[1], q[2], 0.0f);
    }
    __syncthreads();

    const int d = blockIdx.x * 256 + t;           // dst point (one per thread)
    const float px = pos_skip[(size_t)d * 3 + 0];
    const float py = pos_skip[(size_t)d * 3 + 1];
    const float pz = pos_skip[(size_t)d * 3 + 2];

    float d0 = 3.0e38f, d1 = 3.0e38f, d2 = 3.0e38f;
    int   i0 = 0, i1 = 0, i2 = 0;

    for (int j = 0; j < NSRC; ++j) {
        float4 s = sp[j];                          // LDS broadcast (same addr all lanes)
        float dx = px - s.x, dy = py - s.y, dz = pz - s.z;
        float dd = dx * dx + dy * dy + dz * dz;
        if (dd < d2) {
            d2 = dd; i2 = j;
            if (d2 < d1) {
                float tf = d1; d1 = d2; d2 = tf; int ti = i1; i1 = i2; i2 = ti;
                if (d1 < d0) {
                    tf = d0; d0 = d1; d1 = tf; ti = i0; i0 = i1; i1 = ti;
                }
            }
        }
    }

    float w0 = 1.0f / fmaxf(d0, 1e-16f);
    float w1 = 1.0f / fmaxf(d1, 1e-16f);
    float w2 = 1.0f / fmaxf(d2, 1e-16f);
    float inv = 1.0f / (w0 + w1 + w2);

    g_w[(size_t)d * 3 + 0] = w0 * inv;
    g_w[(size_t)d * 3 + 1] = w1 * inv;
    g_w[(size_t)d * 3 + 2] = w2 * inv;
    g_idx[(size_t)d * 3 + 0] = srcbase + i0;
    g_idx[(size_t)d * 3 + 1] = srcbase + i1;
    g_idx[(size_t)d * 3 + 2] = srcbase + i2;
}

// ---------------------------------------------------------------------------
// Kernel 2: fused interpolate + concat + MLP via fp32 WMMA.
// grid = MTOT/16 = 4096 blocks of 256 threads (8 waves). Each block owns a
// 16-row M tile and the full N=256 output width (2 N-tiles of 16 per wave).
// ---------------------------------------------------------------------------
__global__ __launch_bounds__(256) void fp_mlp_kernel(
    const float* __restrict__ x,       // [B*NSRC, 256]
    const float* __restrict__ x_skip,  // [B*NDST, 64]
    const int*   __restrict__ g_idx,   // [MTOT, 3]
    const float* __restrict__ g_w,     // [MTOT, 3]
    const float* __restrict__ W1,      // [320, 256] row-major
    const float* __restrict__ b1,      // [256]
    const float* __restrict__ W2,      // [256, 256]
    const float* __restrict__ b2,      // [256]
    float*       __restrict__ out)     // [MTOT, 256]
{
    __shared__ float ldsA[16 * LDA];   // A0 tile: [16][320] interpolated||skip (+pad)
    __shared__ float ldsH[16 * LDH];   // hidden tile: [16][256] (+pad)
    __shared__ int   sIdx[16 * 3];
    __shared__ float sW[16 * 3];

    const int t     = threadIdx.x;
    const int mbase = blockIdx.x * 16;

    // --- stage knn indices/weights for this tile ---
    if (t < 48) {
        sIdx[t] = g_idx[(size_t)mbase * 3 + t];
        sW[t]   = g_w[(size_t)mbase * 3 + t];
    }
    __syncthreads();

    // --- stage A0 = [knn_interpolate(x) | x_skip] into LDS ---
    // y part: thread t owns column t of all 16 rows (x row reads coalesced).
    {
        const int c = t;  // 0..255
        #pragma unroll 4
        for (int r = 0; r < 16; ++r) {
            int   j0 = sIdx[r * 3 + 0], j1 = sIdx[r * 3 + 1], j2 = sIdx[r * 3 + 2];
            float w0 = sW[r * 3 + 0],   w1 = sW[r * 3 + 1],   w2 = sW[r * 3 + 2];
            float acc = w0 * x[(size_t)j0 * CH + c]
                      + w1 * x[(size_t)j1 * CH + c]
                      + w2 * x[(size_t)j2 * CH + c];
            ldsA[r * LDA + c] = acc;
        }
    }
    // skip part: 16x64
    for (int e = t; e < 16 * CSKIP; e += 256) {
        int r = e >> 6, c = e & 63;
        ldsA[r * LDA + CH + c] = x_skip[(size_t)(mbase + r) * CSKIP + c];
    }
    __syncthreads();

    // --- wave/lane decomposition for WMMA ---
    const int wave = t >> 5;
    const int lane = t & 31;
    const int half = lane >> 4;     // 0: lanes 0-15 (K 0..1), 1: lanes 16-31 (K 2..3)
    const int l16  = lane & 15;     // N offset within tile / M row for A reads
    const int n0   = wave * 32;     // this wave's 32 output columns (two 16-wide tiles)

    // ---------------- GEMM1: hidden = relu(A0 @ W1 + b1), K=320 ----------------
    {
        v8f c0 = {}; v8f c1 = {};
        #pragma unroll 4
        for (int k = 0; k < K1; k += 4) {
            const int krow = k + 2 * half;
            v2f a = { ldsA[l16 * LDA + krow], ldsA[l16 * LDA + krow + 1] };
            v2f bA = { W1[(size_t)krow * NOUT + n0 + l16],
                       W1[(size_t)(krow + 1) * NOUT + n0 + l16] };
            v2f bB = { W1[(size_t)krow * NOUT + n0 + 16 + l16],
                       W1[(size_t)(krow + 1) * NOUT + n0 + 16 + l16] };
            c0 = wmma_f32_4(a, bA, c0);
            c1 = wmma_f32_4(a, bB, c1);
            if (k + 8 < K1)  // uniform branch; stream next W1 K-slice into cache
                __builtin_prefetch(&W1[(size_t)(k + 8 + 2 * half) * NOUT + n0 + l16], 0, 1);
        }
        const float bias0 = b1[n0 + l16];
        const float bias1 = b1[n0 + 16 + l16];
        #pragma unroll
        for (int vv = 0; vv < 8; ++vv) {
            const int r = vv + 8 * half;
            ldsH[r * LDH + n0 + l16]      = fmaxf(c0[vv] + bias0, 0.0f);
            ldsH[r * LDH + n0 + 16 + l16] = fmaxf(c1[vv] + bias1, 0.0f);
        }
    }
    __syncthreads();

    // ---------------- GEMM2: out = relu(H @ W2 + b2), K=256 ----------------
    {
        v8f c0 = {}; v8f c1 = {};
        #pragma unroll 4
        for (int k = 0; k < NOUT; k += 4) {
            const int krow = k + 2 * half;
            v2f a = { ldsH[l16 * LDH + krow], ldsH[l16 * LDH + krow + 1] };
            v2f bA = { W2[(size_t)krow * NOUT + n0 + l16],
                       W2[(size_t)(krow + 1) * NOUT + n0 + l16] };
            v2f bB = { W2[(size_t)krow * NOUT + n0 + 16 + l16],
                       W2[(size_t)(krow + 1) * NOUT + n0 + 16 + l16] };
            c0 = wmma_f32_4(a, bA, c0);
            c1 = wmma_f32_4(a, bB, c1);
            if (k + 8 < NOUT)
                __builtin_prefetch(&W2[(size_t)(k + 8 + 2 * half) * NOUT + n0 + l16], 0, 1);
        }
        const float bias0 = b2[n0 + l16];
        const float bias1 = b2[n0 + 16 + l16];
        #pragma unroll
        for (int vv = 0; vv < 8; ++vv) {
            const int r = vv + 8 * half;
            out[(size_t)(mbase + r) * NOUT + n0 + l16]      = fmaxf(c0[vv] + bias0, 0.0f);
            out[(size_t)(mbase + r) * NOUT + n0 + 16 + l16] = fmaxf(c1[vv] + bias1, 0.0f);
        }
    }
}

// ---------------------------------------------------------------------------
// Launch
// ---------------------------------------------------------------------------
extern "C" void kernel_launch(void* const* d_in, const int* in_sizes, int n_in,
                              void* d_out, int out_size, void* d_ws, size_t ws_size,
                              hipStream_t stream) {
    const float* x         = (const float*)d_in[0];
    const float* pos       = (const float*)d_in[1];
    // d_in[2] = batch (unused: contiguous equal-size clouds, as in reference)
    const float* x_skip    = (const float*)d_in[3];
    const float* pos_skip  = (const float*)d_in[4];
    const int*   batch_skip= (const int*)  d_in[5];
    const float* W1        = (const float*)d_in[6];
    const float* b1        = (const float*)d_in[7];
    const float* W2        = (const float*)d_in[8];
    const float* b2        = (const float*)d_in[9];
    float* out = (float*)d_out;

    int*   g_idx = (int*)d_ws;                                        // MTOT*3 ints
    float* g_w   = (float*)((char*)d_ws + (size_t)MTOT * 3 * sizeof(int));

    fp_knn_kernel<<<MTOT / 256, 256, 0, stream>>>(pos, pos_skip, g_idx, g_w);
    fp_mlp_kernel<<<MTOT / 16, 256, 0, stream>>>(x, x_skip, g_idx, g_w,
                                                 W1, b1, W2, b2, out);

    // Tuple outputs 2 and 3 are pass-throughs: copy after h in return order.
    hipMemcpyAsync(out + (size_t)MTOT * NOUT, pos_skip,
                   (size_t)MTOT * 3 * sizeof(float), hipMemcpyDeviceToDevice, stream);
    hipMemcpyAsync(out + (size_t)MTOT * (NOUT + 3), batch_skip,
                   (size_t)MTOT * sizeof(int), hipMemcpyDeviceToDevice, stream);
}